// GraphAttention_29162827940510
// MI455X (gfx1250) — compile-verified
//
#include <hip/hip_runtime.h>
#include <cmath>

// GAT layer for MI455X (gfx1250, wave32).
// Pipeline:
//   K1 gat_gemm_wmma  : h[k] = X @ W[k] + b[k]   (V_WMMA_F32_16X16X4_F32)
//   K2 gat_scores     : s_self = h.a_s, s_nbr = h.a_n  (wave-per-node shuffle reduce)
//   K3 gat_init       : zero out, denom; m = -inf
//   K4 gat_edge_max   : segment max via int-punned float atomicMax
//   K5 gat_edge_exp   : ex = exp(e-m), segment sum via atomicAdd, stash ex
//   K6 gat_aggregate  : out[v] += (ex/denom) * h[u]   (wave per edge-head, coalesced atomics)

#define DIN   128
#define DOUT  64
#define NHEAD 4
#define NEG_SLOPE 0.01f

// LDS image of W: K-pair interleaved so a B fragment (elements K, K+1 of one
// column) is one contiguous 8B v2f -> single ds_load_b64 into the WMMA source
// pair. Row stride 160 floats (== 32 banks mod 64) makes the two lane-halves
// (rows kk/2 and kk/2+1) hit disjoint bank sets -> conflict-free b64 reads.
#define WROW  (2 * DOUT + 32)   // 160 floats per K-pair row

typedef __attribute__((ext_vector_type(2))) float v2f;
typedef __attribute__((ext_vector_type(8))) float v8f;

// ---------------- Kernel 1: WMMA GEMM  h[k] = X @ W[k] + b[k] ----------------
// Block: 256 threads (8 waves). Grid: (ceil(N/128), NHEAD).
// Each wave computes a 16x64 strip of h for one head: 4 N-tiles x (DIN/4) WMMAs.
__global__ __launch_bounds__(256) void gat_gemm_wmma(
    const float* __restrict__ X,    // [N, DIN]
    const float* __restrict__ W,    // [NHEAD, DIN, DOUT]
    const float* __restrict__ bias, // [NHEAD, DOUT]
    float* __restrict__ Hmat,       // [NHEAD, N, DOUT]
    int N)
{
    __shared__ float Wl[(DIN / 2) * WROW];  // 40 KB, K-pair interleaved
    __shared__ float bl[DOUT];

    const int k = blockIdx.y;
    // Stage W[k]: element (kpos, col) -> Wl[(kpos>>1)*WROW + col*2 + (kpos&1)]
    for (int i = threadIdx.x; i < DIN * DOUT; i += 256) {
        const int kpos = i / DOUT;
        const int col  = i - kpos * DOUT;
        Wl[(kpos >> 1) * WROW + col * 2 + (kpos & 1)] = W[(size_t)k * DIN * DOUT + i];
    }
    if (threadIdx.x < DOUT)
        bl[threadIdx.x] = bias[k * DOUT + threadIdx.x];
    __syncthreads();

    const int wave = threadIdx.x >> 5;
    const int lane = threadIdx.x & 31;
    const int lo   = lane & 15;   // column / M index within half
    const int hi   = lane >> 4;   // which half of the wave
    const int m0   = blockIdx.x * 128 + wave * 16;

    // A fragment rows: lane lo holds row m0+lo (clamped for tail; stores guarded).
    int rowA = m0 + lo;
    if (rowA > N - 1) rowA = N - 1;
    const float* __restrict__ xrow = X + (size_t)rowA * DIN;

    v8f acc[4] = {v8f{}, v8f{}, v8f{}, v8f{}};

    #pragma unroll
    for (int kk = 0; kk < DIN; kk += 4) {
        // A 16x4 fp32 layout: VGPR0 -> K = kk + 2*hi ; VGPR1 -> K = kk + 2*hi + 1
        // (contiguous in memory -> one global_load_b64 per lane)
        v2f a = *(const v2f*)&xrow[kk + 2 * hi];
        const float* __restrict__ wrow = &Wl[((kk >> 1) + hi) * WROW];
        #pragma unroll
        for (int nt = 0; nt < 4; ++nt) {
            // B 4x16 fp32 fragment: one contiguous ds_load_b64
            v2f b = *(const v2f*)&wrow[(nt * 16 + lo) * 2];
            acc[nt] = __builtin_amdgcn_wmma_f32_16x16x4_f32(
                false, a, false, b, (short)0, acc[nt], false, false);
        }
    }

    // C/D 16x16 f32 layout: VGPR v, lane -> row = v + 8*hi, col = lo
    #pragma unroll
    for (int nt = 0; nt < 4; ++nt) {
        #pragma unroll
        for (int v = 0; v < 8; ++v) {
            int row = m0 + v + 8 * hi;
            int col = nt * 16 + lo;
            if (row < N)
                Hmat[((size_t)k * N + row) * DOUT + col] = acc[nt][v] + bl[col];
        }
    }
}

// ---------------- Kernel 2: attention scores per (node, head) ----------------
__global__ __launch_bounds__(256) void gat_scores(
    const float* __restrict__ Hmat,   // [NHEAD, N, DOUT]
    const float* __restrict__ a_self, // [NHEAD, DOUT]
    const float* __restrict__ a_nbr,  // [NHEAD, DOUT]
    float* __restrict__ s_self,       // [N, NHEAD]
    float* __restrict__ s_nbr,        // [N, NHEAD]
    int N)
{
    const int gw   = (blockIdx.x * blockDim.x + threadIdx.x) >> 5;  // wave id
    const int lane = threadIdx.x & 31;
    if (gw >= N * NHEAD) return;                 // wave-uniform exit
    const int k = gw % NHEAD;
    const int n = gw / NHEAD;

    const float* __restrict__ hrow = Hmat + ((size_t)k * N + n) * DOUT;
    float h0 = hrow[lane], h1 = hrow[lane + 32];
    float ss = h0 * a_self[k * DOUT + lane] + h1 * a_self[k * DOUT + lane + 32];
    float sn = h0 * a_nbr [k * DOUT + lane] + h1 * a_nbr [k * DOUT + lane + 32];
    #pragma unroll
    for (int off = 16; off > 0; off >>= 1) {
        ss += __shfl_xor(ss, off, 32);
        sn += __shfl_xor(sn, off, 32);
    }
    if (lane == 0) {
        s_self[(size_t)n * NHEAD + k] = ss;
        s_nbr [(size_t)n * NHEAD + k] = sn;
    }
}

// ---------------- Kernel 3: init accumulators ----------------
__global__ void gat_init(float* __restrict__ out, float* __restrict__ mbuf,
                         float* __restrict__ denom, int outN, int nodeN)
{
    const int t = blockIdx.x * blockDim.x + threadIdx.x;
    if (t < outN)  out[t] = 0.0f;
    if (t < nodeN) { mbuf[t] = -INFINITY; denom[t] = 0.0f; }
}

// int-punned float atomic max (always-compiling, correct incl. -inf init)
__device__ __forceinline__ void atomicMaxFloat(float* addr, float val)
{
    if (!(val < 0.0f))
        atomicMax((int*)addr, __float_as_int(val));
    else
        atomicMin((unsigned int*)addr, (unsigned int)__float_as_int(val));
}

__device__ __forceinline__ float leaky(float x)
{
    return x > 0.0f ? x : NEG_SLOPE * x;
}

// ---------------- Kernel 4: segment max over incoming edges ----------------
__global__ void gat_edge_max(const int* __restrict__ esrc, const int* __restrict__ edst,
                             const float* __restrict__ s_self, const float* __restrict__ s_nbr,
                             const float* __restrict__ ab, float* __restrict__ mbuf, int E)
{
    const int t = blockIdx.x * blockDim.x + threadIdx.x;
    if (t >= E * NHEAD) return;
    const int k = t & (NHEAD - 1);
    const int e = t >> 2;
    const int u = esrc[e], v = edst[e];
    float ev = leaky(s_self[(size_t)v * NHEAD + k] + s_nbr[(size_t)u * NHEAD + k] + ab[k]);
    atomicMaxFloat(&mbuf[(size_t)v * NHEAD + k], ev);
}

// ---------------- Kernel 5: ex = exp(e - m); denom += ex ----------------
__global__ void gat_edge_exp(const int* __restrict__ esrc, const int* __restrict__ edst,
                             const float* __restrict__ s_self, const float* __restrict__ s_nbr,
                             const float* __restrict__ ab, const float* __restrict__ mbuf,
                             float* __restrict__ exbuf, float* __restrict__ denom, int E)
{
    const int t = blockIdx.x * blockDim.x + threadIdx.x;
    if (t >= E * NHEAD) return;
    const int k = t & (NHEAD - 1);
    const int e = t >> 2;
    const int u = esrc[e], v = edst[e];
    float ev = leaky(s_self[(size_t)v * NHEAD + k] + s_nbr[(size_t)u * NHEAD + k] + ab[k]);
    float ex = expf(ev - mbuf[(size_t)v * NHEAD + k]);
    exbuf[t] = ex;
    atomicAdd(&denom[(size_t)v * NHEAD + k], ex);
}

// ---------------- Kernel 6: out[v] += (ex/denom) * h[u], wave per (edge, head) ----------------
__global__ __launch_bounds__(256) void gat_aggregate(
    const int* __restrict__ esrc, const int* __restrict__ edst,
    const float* __restrict__ exbuf, const float* __restrict__ denom,
    const float* __restrict__ Hmat, float* __restrict__ out, int N, int E)
{
    const int gw   = (blockIdx.x * blockDim.x + threadIdx.x) >> 5;
    const int lane = threadIdx.x & 31;
    if (gw >= E * NHEAD) return;                 // wave-uniform exit
    const int k = gw & (NHEAD - 1);
    const int e = gw >> 2;
    const int u = esrc[e], v = edst[e];
    const float w = exbuf[gw] / denom[(size_t)v * NHEAD + k];

    const float* __restrict__ hrow = Hmat + ((size_t)k * N + u) * DOUT;  // L2-resident gather
    float* __restrict__ orow = out + ((size_t)k * N + v) * DOUT;
    atomicAdd(&orow[lane],      w * hrow[lane]);
    atomicAdd(&orow[lane + 32], w * hrow[lane + 32]);
}

extern "C" void kernel_launch(void* const* d_in, const int* in_sizes, int n_in,
                              void* d_out, int out_size, void* d_ws, size_t ws_size,
                              hipStream_t stream)
{
    const float* X      = (const float*)d_in[0];
    const int*   esrc   = (const int*)  d_in[1];
    const int*   edst   = (const int*)  d_in[2];
    const float* W      = (const float*)d_in[3];
    const float* bfc    = (const float*)d_in[4];
    const float* a_self = (const float*)d_in[5];
    const float* a_nbr  = (const float*)d_in[6];
    const float* a_b    = (const float*)d_in[7];
    float*       out    = (float*)d_out;

    const int N = in_sizes[0] / DIN;   // 50000
    const int E = in_sizes[1];         // 1600000

    // Workspace layout (floats): h | s_self | s_nbr | m | denom | ex  (~80 MB)
    float* ws     = (float*)d_ws;
    float* Hmat   = ws;                                   // NHEAD*N*DOUT
    float* s_self = Hmat   + (size_t)NHEAD * N * DOUT;
    float* s_nbr  = s_self + (size_t)N * NHEAD;
    float* mbuf   = s_nbr  + (size_t)N * NHEAD;
    float* denom  = mbuf   + (size_t)N * NHEAD;
    float* exbuf  = denom  + (size_t)N * NHEAD;           // E*NHEAD

    // K1: projection GEMM (WMMA)
    dim3 g1((N + 127) / 128, NHEAD);
    gat_gemm_wmma<<<g1, 256, 0, stream>>>(X, W, bfc, Hmat, N);

    // K2: per-(node, head) attention scores
    const int nw = N * NHEAD;
    gat_scores<<<(nw * 32 + 255) / 256, 256, 0, stream>>>(Hmat, a_self, a_nbr, s_self, s_nbr, N);

    // K3: init out / max / denom
    const int outN = NHEAD * N * DOUT;
    gat_init<<<(outN + 255) / 256, 256, 0, stream>>>(out, mbuf, denom, outN, nw);

    // K4/K5: segment softmax over incoming edges
    const int eh = E * NHEAD;
    gat_edge_max<<<(eh + 255) / 256, 256, 0, stream>>>(esrc, edst, s_self, s_nbr, a_b, mbuf, E);
    gat_edge_exp<<<(eh + 255) / 256, 256, 0, stream>>>(esrc, edst, s_self, s_nbr, a_b, mbuf,
                                                       exbuf, denom, E);

    // K6: weighted scatter-aggregate, one wave per (edge, head)
    const long long aggThreads = (long long)eh * 32;
    gat_aggregate<<<(int)((aggThreads + 255) / 256), 256, 0, stream>>>(
        esrc, edst, exbuf, denom, Hmat, out, N, E);
}